// NeighborhoodAttention_46643344835220
// MI455X (gfx1250) — compile-verified
//
#include <hip/hip_runtime.h>

// Shapes (fixed by the reference)
#define BB   4
#define NN   2048
#define CC   512
#define HH   8
#define HD   64
#define WW   13
#define PADW 6
#define NP   2060            // N + 2*pad
#define C3   (3 * CC)        // 1536

typedef __attribute__((ext_vector_type(2))) float v2f;
typedef __attribute__((ext_vector_type(8))) float v8f;

// D = A(16x4) * B(4x16) + C, fp32 in / fp32 accumulate (CDNA5 V_WMMA_F32_16X16X4_F32)
__device__ __forceinline__ v8f wmma_f32_16x16x4(v2f a, v2f b, v8f c) {
  return __builtin_amdgcn_wmma_f32_16x16x4_f32(
      /*neg_a=*/false, a, /*neg_b=*/false, b,
      /*c_mod=*/(short)0, c, /*reuse_a=*/false, /*reuse_b=*/false);
}

// ---------------------------------------------------------------------------
// Kernel 1: qkv = pad(x) @ qkv_w^T + qkv_b, scattered into Q/K/V (B,H,N_,hd),
//           Q pre-scaled by hd^-0.5.
// One wave computes a 16x64 tile (4 N-sub-tiles): A fragment reused 4x,
// 4 independent WMMA chains. Padding handled by clamped-pointer + 0/1 mask
// (no exec divergence in the inner loop).
// ---------------------------------------------------------------------------
__global__ void __launch_bounds__(128) qkv_gemm_kernel(
    const float* __restrict__ x, const float* __restrict__ qkv_w,
    const float* __restrict__ qkv_b, float* __restrict__ Q,
    float* __restrict__ Kq, float* __restrict__ V) {
  const int lane = threadIdx.x;
  const int lo = lane & 15;
  const int hi = lane >> 4;
  const int mtile = blockIdx.x;                      // 0..514
  const int wcol  = blockIdx.y * 4 + threadIdx.y;    // 0..23 (each = 4 n-tiles)
  const int nt0   = wcol * 4;

  // A operand: row (mtile*16+lo), k = kk + 2*hi + {0,1}; clamp + mask padding
  const int m  = mtile * 16 + lo;                    // < 8240 always
  const int bi = m / NP;
  const int np = m - bi * NP;
  const bool mvalid = (np >= PADW) && (np < NN + PADW);
  const int  nx  = mvalid ? (np - PADW) : 0;
  const float msk = mvalid ? 1.0f : 0.0f;
  const float* xrow = x + ((size_t)bi * NN + nx) * CC;

  // B operand: 4 sub-tiles, columns nt0*16+lo + {0,16,32,48}
  const float* wrow = qkv_w + (size_t)(nt0 * 16 + lo) * CC;

  v8f acc[4] = {};
#pragma unroll 4
  for (int kk = 0; kk < CC; kk += 4) {
    const int k0 = kk + 2 * hi;
    v2f a = *(const v2f*)(xrow + k0);
    a[0] *= msk;
    a[1] *= msk;
    const v2f b0 = *(const v2f*)(wrow + k0);
    const v2f b1 = *(const v2f*)(wrow + 16 * CC + k0);
    const v2f b2 = *(const v2f*)(wrow + 32 * CC + k0);
    const v2f b3 = *(const v2f*)(wrow + 48 * CC + k0);
    acc[0] = wmma_f32_16x16x4(a, b0, acc[0]);
    acc[1] = wmma_f32_16x16x4(a, b1, acc[1]);
    acc[2] = wmma_f32_16x16x4(a, b2, acc[2]);
    acc[3] = wmma_f32_16x16x4(a, b3, acc[3]);
  }

  // Epilogue: bias + scatter into (B,H,N_,hd) with q-scale fused.
#pragma unroll
  for (int t = 0; t < 4; ++t) {
    const int jg = (nt0 + t) * 16 + lo;
    const float bias  = qkv_b[jg];
    const int   which = jg >> 9;          // 0=q,1=k,2=v (wave-uniform per tile)
    const int   h     = (jg >> 6) & 7;
    const int   d     = jg & 63;
    float* dstbuf = (which == 0) ? Q : (which == 1 ? Kq : V);
    const float mul = (which == 0) ? 0.125f : 1.0f;  // hd^-0.5
#pragma unroll
    for (int r = 0; r < 8; ++r) {
      const int mg = mtile * 16 + r + 8 * hi;
      const int bo = mg / NP;
      const int no = mg - bo * NP;
      const size_t dst = (((size_t)(bo * HH + h)) * NP + no) * HD + d;
      dstbuf[dst] = (acc[t][r] + bias) * mul;
    }
  }
}

// ---------------------------------------------------------------------------
// Kernel 2: per-window attention. One wave per (b,h,l).
//   S(13x13) = Qwin * Kwin^T  (WMMA, padded to 16x16)
//   attn = softmax_row(S + pos); s[u] = sum_w attn[w][u]
//   out[d] = sum_u s[u] * V[l+u][d]  -> stored as (B,H,hd,L) flat
// ---------------------------------------------------------------------------
__global__ void __launch_bounds__(128) na_attn_kernel(
    const float* __restrict__ Q, const float* __restrict__ Km,
    const float* __restrict__ V, const float* __restrict__ pos,
    float* __restrict__ Y) {
  const int lane = threadIdx.x;
  const int lo = lane & 15;
  const int hi = lane >> 4;
  const int g  = blockIdx.x * 4 + threadIdx.y;     // (b*H + h)*L + l
  const int l  = g & (NN - 1);
  const int bh = g >> 11;

  const size_t base = (size_t)bh * NP * HD;
  const float* qr = Q  + base + (size_t)min(l + lo, NP - 1) * HD;  // A row w=lo
  const float* kr = Km + base + (size_t)min(l + lo, NP - 1) * HD;  // B col u=lo
  const float* Vb = V  + base;

  v8f s = {};
#pragma unroll
  for (int kk = 0; kk < HD; kk += 4) {
    const int k0 = kk + 2 * hi;
    const v2f a = *(const v2f*)(qr + k0);
    const v2f b = *(const v2f*)(kr + k0);
    s = wmma_f32_16x16x4(a, b, s);
  }

  // C/D layout: lane holds S[w = r + 8*hi][u = lo] in acc element r.
  const int u = lo;
  float cs = 0.0f;  // column sum s[u]
#pragma unroll
  for (int r = 0; r < 8; ++r) {
    const int w = r + 8 * hi;
    float val = s[r];
    if (w < WW && u < WW) val += pos[w * WW + u];
    float rmax = (u < WW) ? val : -3.0e38f;
#pragma unroll
    for (int sh = 8; sh >= 1; sh >>= 1)
      rmax = fmaxf(rmax, __shfl_xor(rmax, sh, 16));
    const float e = (u < WW) ? __expf(val - rmax) : 0.0f;
    float rsum = e;
#pragma unroll
    for (int sh = 8; sh >= 1; sh >>= 1)
      rsum += __shfl_xor(rsum, sh, 16);
    if (w < WW) cs += e / rsum;
  }
  cs += __shfl_xor(cs, 16, 32);   // combine the two 8-row halves -> full s[u]

  // out[d] = sum_u s[u]*V[l+u][d]; each lane produces d = 2*lane, 2*lane+1
  const int dd = lane * 2;
  float o0 = 0.0f, o1 = 0.0f;
#pragma unroll
  for (int uu = 0; uu < WW; ++uu) {
    const float su = __shfl(cs, uu, 16);
    const v2f vv = *(const v2f*)(Vb + (size_t)(l + uu) * HD + dd);
    o0 += su * vv[0];
    o1 += su * vv[1];
  }

  // Store in reference's transpose(0,1,3,2).reshape(B,N,C) order: (B,H,hd,L)
  float* yrow = Y + ((size_t)bh * HD + dd) * NN + l;
  yrow[0]  = o0;
  yrow[NN] = o1;
}

// ---------------------------------------------------------------------------
// Kernel 3: out = Y @ proj_w^T + proj_b. One wave per 16x64 tile
// (4 N-sub-tiles, same register blocking as kernel 1).
// ---------------------------------------------------------------------------
__global__ void __launch_bounds__(128) proj_gemm_kernel(
    const float* __restrict__ Yin, const float* __restrict__ proj_w,
    const float* __restrict__ proj_b, float* __restrict__ out) {
  const int lane = threadIdx.x;
  const int lo = lane & 15;
  const int hi = lane >> 4;
  const int mtile = blockIdx.x;                      // 0..511 (M = B*N = 8192)
  const int wcol  = blockIdx.y * 4 + threadIdx.y;    // 0..7 (each = 4 n-tiles)
  const int nt0   = wcol * 4;

  const float* yrow = Yin + (size_t)(mtile * 16 + lo) * CC;
  const float* wrow = proj_w + (size_t)(nt0 * 16 + lo) * CC;

  v8f acc[4] = {};
#pragma unroll 4
  for (int kk = 0; kk < CC; kk += 4) {
    const int k0 = kk + 2 * hi;
    const v2f a  = *(const v2f*)(yrow + k0);
    const v2f b0 = *(const v2f*)(wrow + k0);
    const v2f b1 = *(const v2f*)(wrow + 16 * CC + k0);
    const v2f b2 = *(const v2f*)(wrow + 32 * CC + k0);
    const v2f b3 = *(const v2f*)(wrow + 48 * CC + k0);
    acc[0] = wmma_f32_16x16x4(a, b0, acc[0]);
    acc[1] = wmma_f32_16x16x4(a, b1, acc[1]);
    acc[2] = wmma_f32_16x16x4(a, b2, acc[2]);
    acc[3] = wmma_f32_16x16x4(a, b3, acc[3]);
  }

#pragma unroll
  for (int t = 0; t < 4; ++t) {
    const int jg = (nt0 + t) * 16 + lo;
    const float bias = proj_b[jg];
#pragma unroll
    for (int r = 0; r < 8; ++r) {
      const int mg = mtile * 16 + r + 8 * hi;
      out[(size_t)mg * CC + jg] = acc[t][r] + bias;
    }
  }
}

// ---------------------------------------------------------------------------
extern "C" void kernel_launch(void* const* d_in, const int* in_sizes, int n_in,
                              void* d_out, int out_size, void* d_ws, size_t ws_size,
                              hipStream_t stream) {
  const float* x      = (const float*)d_in[0];
  const float* qkv_w  = (const float*)d_in[1];
  const float* qkv_b  = (const float*)d_in[2];
  const float* pos    = (const float*)d_in[3];
  const float* proj_w = (const float*)d_in[4];
  const float* proj_b = (const float*)d_in[5];
  float* out = (float*)d_out;
  float* ws  = (float*)d_ws;

  const size_t qkv_elems = (size_t)BB * HH * NP * HD;  // 4,218,880 per tensor
  float* Q = ws;
  float* K = ws + qkv_elems;
  float* V = ws + 2 * qkv_elems;
  float* Y = ws + 3 * qkv_elems;                       // B*N*C = 4,194,304

  const dim3 blk(32, 4, 1);
  // 8240/16 = 515 M-tiles; 1536 cols = 24 wave-columns of 64 (4 per block)
  qkv_gemm_kernel<<<dim3(515, 6, 1), blk, 0, stream>>>(x, qkv_w, qkv_b, Q, K, V);
  // B*H*L = 65536 windows, 4 per block
  na_attn_kernel<<<dim3(16384, 1, 1), blk, 0, stream>>>(Q, K, V, pos, Y);
  // 8192/16 = 512 M-tiles; 512 cols = 8 wave-columns of 64 (4 per block)
  proj_gemm_kernel<<<dim3(512, 2, 1), blk, 0, stream>>>(Y, proj_w, proj_b, out);
}